// ContrastMemory_15685220565754
// MI455X (gfx1250) — compile-verified
//
#include <hip/hip_runtime.h>
#include <hip/hip_bf16.h>

#define BATCH_N   256
#define FDIM      128
#define KNEG      4096
#define KP1       4097
#define CAPACITY  500000
#define MOMENTUM  0.5f

// Type matching the builtin's parameter: int __attribute__((vector_size(16)))
typedef int v4i __attribute__((vector_size(16)));
typedef __attribute__((address_space(1))) v4i* gptr_v4i;   // global
typedef __attribute__((address_space(3))) v4i* lptr_v4i;   // LDS

#if __has_builtin(__builtin_amdgcn_global_load_async_to_lds_b128) && \
    __has_builtin(__builtin_amdgcn_global_store_async_from_lds_b128)
#define HAVE_ASYNC_LDS 1
#else
#define HAVE_ASYNC_LDS 0
#endif

#if __has_builtin(__builtin_amdgcn_s_wait_asynccnt)
#define WAIT_ASYNC0() __builtin_amdgcn_s_wait_asynccnt(0)
#else
#define WAIT_ASYNC0() asm volatile("s_wait_asynccnt 0x0" ::: "memory")
#endif

// gfx12-style CPol temporal hints
#define CPOL_RT 0   // regular temporal (loads of the reused memory tables)
#define CPOL_NT 1   // non-temporal (write-once streaming output)

// ---------------------------------------------------------------------------
// Kernel 3 (placed first so its disassembly is visible in the snippet):
// the big gather. One wave32 per 512B output row; grid-stride over
// 2 * 256 * 4097 rows. Row copy goes global -> LDS -> global through the
// CDNA5 async-DMA path (ASYNCcnt), double-buffered per wave.
// ---------------------------------------------------------------------------
__global__ void __launch_bounds__(256)
cm_gather_kernel(const float* __restrict__ mem1,
                 const float* __restrict__ mem2,
                 const int*   __restrict__ s_layer_p,
                 const int*   __restrict__ t_layer_p,
                 const int*   __restrict__ idx,
                 const int*   __restrict__ cidx,
                 const float* __restrict__ upd_s,
                 const float* __restrict__ upd_t,
                 const int*   __restrict__ map,
                 float* __restrict__ out) {
    __shared__ float lds[(256 / 32) * 2 * FDIM];   // 8 waves x 2 buffers x 128 f32 = 8 KB

    const int lane = threadIdx.x & 31;
    const int wib  = threadIdx.x >> 5;             // wave-in-block
    const int nwaves = gridDim.x * (blockDim.x >> 5);
    int       wave   = blockIdx.x * (blockDim.x >> 5) + wib;

    const int   sl = s_layer_p[0];
    const int   tl = t_layer_p[0];
    const float* base_s = mem1 + (size_t)sl * CAPACITY * FDIM;
    const float* base_t = mem2 + (size_t)tl * CAPACITY * FDIM;

    const int PER_SIDE = BATCH_N * KP1;            // 1,048,832
    const int NROWS    = 2 * PER_SIDE;             // 2,097,664

    float* ldsw = lds + wib * 2 * FDIM;
    int buf = 0;

    for (int rid = wave; rid < NROWS; rid += nwaves) {
        const int side = rid / PER_SIDE;
        const int rem  = rid - side * PER_SIDE;
        const int b    = rem / KP1;
        const int j    = rem - b * KP1;

        const int r = (j == 0) ? idx[b] : cidx[(size_t)b * KNEG + (j - 1)];
        const int t = map[r];

        const float* src;
        if (t >= 0) src = (side ? upd_t  : upd_s)  + (size_t)t * FDIM;
        else        src = (side ? base_t : base_s) + (size_t)r * FDIM;

        float* dst = out + (size_t)rid * FDIM;

#if HAVE_ASYNC_LDS
        float* lbuf = ldsw + buf * FDIM;
        // gather 512B row into LDS (32 lanes x 16B), no VGPR data movement
        __builtin_amdgcn_global_load_async_to_lds_b128(
            (gptr_v4i)(const_cast<float*>(src) + lane * 4),
            (lptr_v4i)(lbuf + lane * 4), 0, CPOL_RT);
        WAIT_ASYNC0();   // load complete (and previous store drained)
        // stream it out non-temporally: output is write-once, keep L2 for the tables
        __builtin_amdgcn_global_store_async_from_lds_b128(
            (gptr_v4i)(dst + lane * 4),
            (lptr_v4i)(lbuf + lane * 4), 0, CPOL_NT);
        buf ^= 1;
#else
        float4 v = ((const float4*)src)[lane];
        ((float4*)dst)[lane] = v;
#endif
    }
#if HAVE_ASYNC_LDS
    WAIT_ASYNC0();
#endif
}

// ---------------------------------------------------------------------------
// Kernel 1: reset the row->slot map (deterministic across replays)
// ---------------------------------------------------------------------------
__global__ void cm_init_map_kernel(int* __restrict__ map) {
    int i = blockIdx.x * blockDim.x + threadIdx.x;
    if (i < CAPACITY) map[i] = -1;
}

// ---------------------------------------------------------------------------
// Kernel 2: EMA update + L2 normalize the touched rows into workspace.
// One wave32 per (side, b): 32 lanes x float4 = 128 floats.
// Grid: 64 blocks x 256 threads = 512 waves (256 for v1/s, 256 for v2/t).
// ---------------------------------------------------------------------------
__global__ void __launch_bounds__(256)
cm_ema_update_kernel(const float* __restrict__ f_s,
                     const float* __restrict__ f_t,
                     const float* __restrict__ mem1,
                     const float* __restrict__ mem2,
                     const int*   __restrict__ s_layer_p,
                     const int*   __restrict__ t_layer_p,
                     const int*   __restrict__ idx,
                     float* __restrict__ upd_s,
                     float* __restrict__ upd_t,
                     int*   __restrict__ map) {
    const int tid   = blockIdx.x * blockDim.x + threadIdx.x;
    const int wave  = tid >> 5;          // 0..511
    const int lane  = tid & 31;
    const int side  = wave >> 8;         // 0 = student (v1), 1 = teacher (v2)
    const int b     = wave & 255;

    const float* f     = side ? f_t  : f_s;
    const float* mem   = side ? mem2 : mem1;
    const int    layer = side ? t_layer_p[0] : s_layer_p[0];
    float*       upd   = side ? upd_t : upd_s;

    const int r = idx[b];
    const float4* src = (const float4*)(mem + ((size_t)layer * CAPACITY + (size_t)r) * FDIM);
    const float4* fp  = (const float4*)(f + (size_t)b * FDIM);

    float4 o = src[lane];
    float4 x = fp[lane];
    float4 n;
    n.x = o.x * MOMENTUM + x.x * (1.0f - MOMENTUM);
    n.y = o.y * MOMENTUM + x.y * (1.0f - MOMENTUM);
    n.z = o.z * MOMENTUM + x.z * (1.0f - MOMENTUM);
    n.w = o.w * MOMENTUM + x.w * (1.0f - MOMENTUM);

    float ss = n.x * n.x + n.y * n.y + n.z * n.z + n.w * n.w;
#pragma unroll
    for (int off = 16; off > 0; off >>= 1)
        ss += __shfl_xor(ss, off, 32);

    const float inv = 1.0f / sqrtf(ss);
    n.x *= inv; n.y *= inv; n.z *= inv; n.w *= inv;

    ((float4*)(upd + (size_t)b * FDIM))[lane] = n;

    if (side == 0 && lane == 0) map[r] = b;   // same idx for both memories
}

// ---------------------------------------------------------------------------
extern "C" void kernel_launch(void* const* d_in, const int* in_sizes, int n_in,
                              void* d_out, int out_size, void* d_ws, size_t ws_size,
                              hipStream_t stream) {
    const float* f_s  = (const float*)d_in[0];
    const float* f_t  = (const float*)d_in[1];
    const float* mem1 = (const float*)d_in[2];
    const float* mem2 = (const float*)d_in[3];
    const int* s_layer = (const int*)d_in[4];
    const int* t_layer = (const int*)d_in[5];
    const int* idx     = (const int*)d_in[6];
    const int* cidx    = (const int*)d_in[7];
    float* out = (float*)d_out;

    // workspace: upd_s (256x128 f32) | upd_t (256x128 f32) | map (500000 i32)
    float* upd_s = (float*)d_ws;
    float* upd_t = upd_s + BATCH_N * FDIM;
    int*   map   = (int*)(upd_t + BATCH_N * FDIM);

    cm_init_map_kernel<<<(CAPACITY + 255) / 256, 256, 0, stream>>>(map);
    cm_ema_update_kernel<<<64, 256, 0, stream>>>(f_s, f_t, mem1, mem2,
                                                 s_layer, t_layer, idx,
                                                 upd_s, upd_t, map);
    cm_gather_kernel<<<4096, 256, 0, stream>>>(mem1, mem2, s_layer, t_layer,
                                               idx, cidx, upd_s, upd_t, map, out);
}